// ARMPS_13400297963987
// MI455X (gfx1250) — compile-verified
//
#include <hip/hip_runtime.h>
#include <hip/hip_bf16.h>

typedef float v2f __attribute__((ext_vector_type(2)));
typedef float v8f __attribute__((ext_vector_type(8)));

#define N_  256
#define D_  16
#define F_  2
#define BS_ 512

__device__ __forceinline__ v8f wmma4(v2f a, v2f b, v8f c) {
  // D(16x16,f32) = A(16x4,f32) x B(4x16,f32) + C
  return __builtin_amdgcn_wmma_f32_16x16x4_f32(
      /*neg_a=*/false, a, /*neg_b=*/false, b,
      /*c_mod=*/(short)0, c, /*reuse_a=*/false, /*reuse_b=*/false);
}

// Per-batch init term: logp0[b] = log_softmax(tensors[0,0,0,0,:] + 1)[data[b,0]]
__global__ void mps_init(const int* __restrict__ data,
                         const float* __restrict__ tensors,
                         float* __restrict__ out) {
  int b = blockIdx.x * blockDim.x + threadIdx.x;
  if (b >= BS_) return;
  float l0 = tensors[0] + 1.0f;
  float l1 = tensors[1] + 1.0f;
  float mx = fmaxf(l0, l1);
  float lse = mx + __logf(__expf(l0 - mx) + __expf(l1 - mx));
  int x = data[(size_t)b * N_ + 0];
  out[b] = (x ? l1 : l0) - lse;
}

// One-shot packing of per-batch-row selector bits: selg[tile][j] bit m = data[tile*16+m, j]
__global__ void __launch_bounds__(32) mps_selpack(const int* __restrict__ data,
                                                  unsigned int* __restrict__ selg) {
  const int tid = threadIdx.x;
  const int lo = tid & 15, hi = tid >> 4;
  const int tile = blockIdx.x;
  const size_t row = (size_t)(tile * 16 + lo);
  for (int j0 = 0; j0 < N_; j0 += 2) {
    int v = data[row * N_ + j0 + hi];             // lanes 0-15 -> j0, 16-31 -> j0+1
    unsigned long long bal = __ballot(v != 0);
    if (tid == 0)  selg[tile * N_ + j0]     = (unsigned int)(bal & 0xFFFFu);
    if (tid == 16) selg[tile * N_ + j0 + 1] = (unsigned int)((bal >> 16) & 0xFFFFu);
  }
}

// One wave handles (n, two 16-row batch tiles); the B matrices are shared.
// Update per tile: V <- V (acc init, the +I term) + A*B0raw + Amasked*(B1raw-B0raw)
template <bool HAVE_SEL>
__global__ void __launch_bounds__(256) mps_chain(
    const int* __restrict__ data, const float* __restrict__ tensors,
    const unsigned int* __restrict__ selg,
    float* __restrict__ contrib, float* __restrict__ out, int use_atomic) {
  extern __shared__ float smem[];   // [8][512] tr (+ [8][2][256] sel if !HAVE_SEL)

  const int tid = threadIdx.x & 31;
  const int wv  = __builtin_amdgcn_readfirstlane((int)(threadIdx.x >> 5));
  const int lo = tid & 15, hi = tid >> 4;
  const int tile0 = (blockIdx.x * 8 + wv) * 2;    // 0,2,..,30
  const int n     = blockIdx.y + 1;               // 1..255

  float* tr = smem + wv * 512;                    // wave-private 2x(16x16) transpose buf
  unsigned int* selL = (unsigned int*)(smem + 8 * 512) + wv * 2 * N_;

  if (!HAVE_SEL) {
    // Fallback: build both tiles' selector bits in LDS (wave-private, no block sync).
#pragma unroll
    for (int t = 0; t < 2; ++t) {
      const size_t row = (size_t)((tile0 + t) * 16 + lo);
      for (int j0 = 0; j0 <= n; j0 += 2) {
        int v = data[row * N_ + j0 + hi];
        unsigned long long bal = __ballot(v != 0);
        if (tid == 0)  selL[t * N_ + j0] = (unsigned int)(bal & 0xFFFFu);
        if (tid == 16 && (j0 + 1) < N_)
          selL[t * N_ + j0 + 1] = (unsigned int)((bal >> 16) & 0xFFFFu);
      }
    }
    __builtin_amdgcn_wave_barrier();
  }
  auto getsel = [&](int t, int j) -> unsigned int {
    return HAVE_SEL ? selg[(size_t)(tile0 + t) * N_ + j] : selL[t * N_ + j];
  };

  const size_t tbase = (size_t)n * (N_ * D_ * D_ * F_);   // tensors[n, ...]

  // ---- Init V in C/D layout: vT[i] = V[m=i+8*hi, r=lo] ----
  v8f v0, v1;
  {
    v2f w = *(const v2f*)(tensors + tbase + (size_t)lo * F_);   // j=0,l=0,r=lo,f=0/1
    float addI = (lo == 0) ? 1.0f : 0.0f;
    unsigned int s0 = getsel(0, 0), s1 = getsel(1, 0);
#pragma unroll
    for (int i = 0; i < 8; ++i) {
      int m = i + 8 * hi;
      v0[i] = (((s0 >> m) & 1u) ? w[1] : w[0]) + addI;
      v1[i] = (((s1 >> m) & 1u) ? w[1] : w[0]) + addI;
    }
  }

  const int cstore = hi * 128 + lo;      // (m*16 + r), + i*16
  const int aload  = lo * 16 + 2 * hi;   // (m*16 + k), + 4c (+1)

  auto loadB = [&](int j, v2f b0o[4], v2f b1o[4]) {
    const float* gm = tensors + tbase + (size_t)j * (D_ * D_ * F_);
#pragma unroll
    for (int c = 0; c < 4; ++c) {
      int k0 = 4 * c + 2 * hi;
      b0o[c] = *(const v2f*)(gm + ((size_t)k0 * D_ + lo) * F_);        // f=0/1 packed
      b1o[c] = *(const v2f*)(gm + ((size_t)(k0 + 1) * D_ + lo) * F_);
    }
  };

  // Software-pipelined chain
  v2f b0[4], b1[4];
  unsigned int sj0 = 0, sj1 = 0;
  if (n > 1) { loadB(1, b0, b1); sj0 = getsel(0, 1); sj1 = getsel(1, 1); }

  for (int j = 1; j < n; ++j) {
    // Issue next step's loads first so L2 latency overlaps transpose + WMMA.
    v2f nb0[4], nb1[4];
    unsigned int nsj0 = 0, nsj1 = 0;
    if (j + 1 < n) {
      loadB(j + 1, nb0, nb1);
      nsj0 = getsel(0, j + 1);
      nsj1 = getsel(1, j + 1);
    }

    // C-layout -> A-layout transpose through wave-private LDS (DS is in-order per wave).
#pragma unroll
    for (int i = 0; i < 8; ++i) {
      tr[cstore + i * 16]       = v0[i];
      tr[256 + cstore + i * 16] = v1[i];
    }
    __builtin_amdgcn_wave_barrier();
    v2f a0[4], a1[4];
#pragma unroll
    for (int c = 0; c < 4; ++c) {
      a0[c][0] = tr[aload + 4 * c];
      a0[c][1] = tr[aload + 4 * c + 1];
      a1[c][0] = tr[256 + aload + 4 * c];
      a1[c][1] = tr[256 + aload + 4 * c + 1];
    }
    __builtin_amdgcn_wave_barrier();

    // Per-lane A-row masks: in A layout the row index is lo for both halves.
    const bool sr0 = ((sj0 >> lo) & 1u) != 0u;
    const bool sr1 = ((sj1 >> lo) & 1u) != 0u;

#pragma unroll
    for (int c = 0; c < 4; ++c) {
      v2f bf0; bf0[0] = b0[c][0];            bf0[1] = b1[c][0];            // f=0 raw
      v2f bfd; bfd[0] = b0[c][1] - b0[c][0]; bfd[1] = b1[c][1] - b1[c][0]; // f1-f0 raw
      v2f am0; am0[0] = sr0 ? a0[c][0] : 0.0f; am0[1] = sr0 ? a0[c][1] : 0.0f;
      v2f am1; am1[0] = sr1 ? a1[c][0] : 0.0f; am1[1] = sr1 ? a1[c][1] : 0.0f;
      v0 = wmma4(a0[c], bf0, v0);            // two independent acc chains
      v1 = wmma4(a1[c], bf0, v1);
      v0 = wmma4(am0,  bfd, v0);
      v1 = wmma4(am1,  bfd, v1);
    }

#pragma unroll
    for (int c = 0; c < 4; ++c) { b0[c] = nb0[c]; b1[c] = nb1[c]; }
    sj0 = nsj0; sj1 = nsj1;
  }

  // ---- Step-n logits: logit_f = sum_l V[b,l] * (tensors[n,n,l,0,f] + I[l,0]) ----
  v2f wfin = *(const v2f*)(tensors + tbase + (size_t)n * (D_ * D_ * F_) +
                           ((size_t)lo * D_) * F_);
  float addI = (lo == 0) ? 1.0f : 0.0f;
  float w0 = wfin[0] + addI;
  float w1 = wfin[1] + addI;
#pragma unroll
  for (int t = 0; t < 2; ++t) {
    unsigned int sn = getsel(t, n);
#pragma unroll
    for (int i = 0; i < 8; ++i) {
      float t0 = (t == 0 ? v0[i] : v1[i]) * w0;
      float t1 = (t == 0 ? v0[i] : v1[i]) * w1;
#pragma unroll
      for (int s = 8; s >= 1; s >>= 1) {          // reduce over l within 16-lane halves
        t0 += __shfl_xor(t0, s, 16);
        t1 += __shfl_xor(t1, s, 16);
      }
      if (lo == 0) {
        int m = i + 8 * hi;
        int b = (tile0 + t) * 16 + m;
        float mx  = fmaxf(t0, t1);
        float lse = mx + __logf(__expf(t0 - mx) + __expf(t1 - mx));
        float lx  = ((sn >> m) & 1u) ? t1 : t0;
        float cv  = lx - lse;
        if (use_atomic) atomicAdd(&out[b], cv);
        else            contrib[(size_t)(n - 1) * BS_ + b] = cv;
      }
    }
  }
}

// Deterministic fixed-order reduction over n
__global__ void mps_reduce(const float* __restrict__ contrib,
                           float* __restrict__ out) {
  int b = blockIdx.x * blockDim.x + threadIdx.x;
  if (b >= BS_) return;
  float s = out[b];
  for (int r = 0; r < N_ - 1; ++r) s += contrib[(size_t)r * BS_ + b];
  out[b] = s;
}

extern "C" void kernel_launch(void* const* d_in, const int* in_sizes, int n_in,
                              void* d_out, int out_size, void* d_ws, size_t ws_size,
                              hipStream_t stream) {
  const int*   data    = (const int*)d_in[0];     // (BS, N) selectors in {0,1}
  const float* tensors = (const float*)d_in[1];   // (N, N, D, D, F) f32
  float*       out     = (float*)d_out;           // (BS,) f32

  const size_t sel_bytes     = (size_t)32 * N_ * sizeof(unsigned int);   // 32 KB
  const size_t contrib_bytes = (size_t)(N_ - 1) * BS_ * sizeof(float);   // 510 KB

  const bool have_sel = ws_size >= sel_bytes;
  unsigned int* selg = (unsigned int*)d_ws;
  float* contrib = have_sel ? (float*)((char*)d_ws + sel_bytes) : (float*)d_ws;
  const size_t need = have_sel ? sel_bytes + contrib_bytes : contrib_bytes;
  const int use_atomic = (ws_size < need) ? 1 : 0;

  mps_init<<<dim3((BS_ + 255) / 256), dim3(256), 0, stream>>>(data, tensors, out);

  const size_t smem = 8 * 512 * sizeof(float) +
                      (have_sel ? 0 : 8 * 2 * N_ * sizeof(unsigned int));
  if (have_sel) {
    mps_selpack<<<dim3(32), dim3(32), 0, stream>>>(data, selg);
    mps_chain<true><<<dim3(2, N_ - 1), dim3(256), smem, stream>>>(
        data, tensors, selg, contrib, out, use_atomic);
  } else {
    mps_chain<false><<<dim3(2, N_ - 1), dim3(256), smem, stream>>>(
        data, tensors, selg, contrib, out, use_atomic);
  }
  if (!use_atomic)
    mps_reduce<<<dim3((BS_ + 255) / 256), dim3(256), 0, stream>>>(contrib, out);
}